// DeletionChannel_6296422056027
// MI455X (gfx1250) — compile-verified
//
#include <hip/hip_runtime.h>
#include <math.h>
#include <stdint.h>

#define Bn 128
#define Ln 10
#define Vn 32
#define MINF (-3.4028234663852886e38f)

typedef __attribute__((ext_vector_type(2))) float v2f;
typedef __attribute__((ext_vector_type(8))) float v8f;

__device__ __forceinline__ float wave_max(float v) {
#pragma unroll
    for (int off = 16; off > 0; off >>= 1)
        v = fmaxf(v, __shfl_xor(v, off, 32));
    return v;
}
__device__ __forceinline__ float wave_sum(float v) {
#pragma unroll
    for (int off = 16; off > 0; off >>= 1)
        v += __shfl_xor(v, off, 32);
    return v;
}

// d_ws layout (floats): [0..191] W[16][12] (WMMA A matrix, zero-padded),
//                       [192..312] ndl[11][11]
__global__ __launch_bounds__(32) void dc_init_consts(float* __restrict__ ws) {
    const int lane = threadIdx.x;
    const double lp = log(0.1), l1p = log1p(-0.1);
    for (int t = lane; t < 192; t += 32) ws[t] = 0.f;
    for (int t = lane; t < 121; t += 32) ws[192 + t] = MINF;
    if (lane < 10) {
        const int i = lane;
        const double mx = 10.0 * l1p;  // max combo logit (ndel=0), always unmasked
        double s = 0.0;
        for (int c = 0; c < 1024; ++c) {
            int nd = __popc(c);
            if (nd <= 9 - i) s += exp(lp * nd + l1p * (10 - nd) - mx);
        }
        const double lse = mx + log(s);
        double Wrow[10];
#pragma unroll
        for (int j = 0; j < 10; ++j) Wrow[j] = 0.0;
        for (int c = 0; c < 1024; ++c) {
            int nd = __popc(c);
            if (nd > 9 - i) continue;
            int p[10];
#pragma unroll
            for (int k = 0; k < 10; ++k) p[k] = k;
#pragma unroll
            for (int step = 1; step < 10; ++step) {
                const int idx = 9 - step;
                if ((c >> (9 - idx)) & 1) {
                    int t = p[idx];
#pragma unroll
                    for (int k = idx; k < 9; ++k) p[k] = p[k + 1];
                    p[9] = t;
                }
            }
            double w = exp(lp * nd + l1p * (10 - nd) - lse);
            int pi = p[i];
#pragma unroll
            for (int j = 0; j < 10; ++j) Wrow[j] += (pi == j) ? w : 0.0;
        }
#pragma unroll
        for (int j = 0; j < 10; ++j) ws[i * 12 + j] = (float)Wrow[j];
    }
    if (lane < 11) {
        const int n = lane;
        for (int k = 0; k <= n; ++k) {
            double v = lgamma((double)(n + 1)) - lgamma((double)(k + 1)) -
                       lgamma((double)(n - k + 1)) + k * lp + (n - k) * l1p;
            ws[192 + n * 11 + (n - k)] = (float)v;
        }
    }
}

__global__ __launch_bounds__(32) void dc_main(
    const float* __restrict__ messages,
    const float* __restrict__ logits,
    const uint8_t* __restrict__ mask,
    float* __restrict__ out,
    const float* __restrict__ cst) {
    __shared__ float Qs[12][32];   // B operand: Q[j][v'], zero padded j=10,11 / v'=31
    __shared__ float Ex[16][32];   // expected[i][v']
    __shared__ float mArr[32];
    __shared__ float eosArr[Ln];

    const int b = blockIdx.x;
    const int lane = threadIdx.x;      // = vocab column v
    const int half = lane >> 4;
    const int l15 = lane & 15;
    const float* lg = logits + (size_t)b * Ln * Vn;
    const float* ms = messages + (size_t)b * Ln * Vn;
    const float* Wg = cst;             // [16][12]
    const float* ndl = cst + 192;      // [11][11]

#pragma unroll
    for (int j = 0; j < 12; ++j) Qs[j][lane] = 0.f;
    if (lane == 0) mArr[31] = 0.f;

    float vals[Ln];
#pragma unroll
    for (int i = 0; i < Ln; ++i) vals[i] = lg[i * Vn + lane];
    if (lane == 0) {
#pragma unroll
        for (int i = 0; i < Ln; ++i) eosArr[i] = vals[i];
    }

    // non_eos = log_softmax over v=1..31 per row; lane l holds column l
    float ne[Ln];
#pragma unroll
    for (int i = 0; i < Ln; ++i) {
        float x = (lane == 0) ? MINF : vals[i];
        float mx = wave_max(x);
        float s = wave_sum((lane == 0) ? 0.f : expf(vals[i] - mx));
        float lse = mx + logf(s);
        ne[i] = vals[i] - lse;
    }
    // m[v'] = max_j non_eos[j][v'] is lane-local; Q[j][v'] = exp(ne-m)
    float mV = MINF;
#pragma unroll
    for (int i = 0; i < Ln; ++i) mV = fmaxf(mV, ne[i]);
    if (lane >= 1) {
        mArr[lane - 1] = mV;
#pragma unroll
        for (int i = 0; i < Ln; ++i) Qs[i][lane - 1] = expf(ne[i] - mV);
    }
    __syncthreads();

    // E'(16x32) = W(16x12) @ Q(12x32) via V_WMMA_F32_16X16X4_F32 (2 N-tiles, 3 K-steps)
    v8f acc0 = {};
    v8f acc1 = {};
#pragma unroll
    for (int kc = 0; kc < 3; ++kc) {
        v2f a, b0, b1;
        a.x = Wg[l15 * 12 + 4 * kc + 2 * half];       // A: VGPR0 = K, K+2
        a.y = Wg[l15 * 12 + 4 * kc + 2 * half + 1];   //    VGPR1 = K+1, K+3
        b0.x = Qs[4 * kc + 2 * half][l15];
        b0.y = Qs[4 * kc + 2 * half + 1][l15];
        b1.x = Qs[4 * kc + 2 * half][16 + l15];
        b1.y = Qs[4 * kc + 2 * half + 1][16 + l15];
        acc0 = __builtin_amdgcn_wmma_f32_16x16x4_f32(false, a, false, b0, (short)0, acc0, false, false);
        acc1 = __builtin_amdgcn_wmma_f32_16x16x4_f32(false, a, false, b1, (short)0, acc1, false, false);
    }
    // D layout: VGPR r -> row r (lanes 0-15) / row r+8 (lanes 16-31)
#pragma unroll
    for (int r = 0; r < 8; ++r) {
        const int i = r + 8 * half;
        Ex[i][l15]      = mArr[l15]      + logf(fmaxf(acc0[r], 1e-37f));
        Ex[i][16 + l15] = mArr[16 + l15] + logf(fmaxf(acc1[r], 1e-37f));
    }
    __syncthreads();

    // ---- EOS / length chain (lane-uniform, computed redundantly) ----
    float eosv[Ln], lg1m[Ln];
#pragma unroll
    for (int i = 0; i < Ln; ++i) {
        eosv[i] = eosArr[i];
        lg1m[i] = log1pf(-expf(eosv[i]));
    }
    float llg[Ln + 1];
    llg[0] = eosv[0];
    float csum = 0.f;
#pragma unroll
    for (int i = 1; i < Ln; ++i) {
        csum += lg1m[i - 1];
        llg[i] = eosv[i] + fminf(fmaxf(csum, MINF), 0.f);
    }
    csum += lg1m[Ln - 1];
    llg[Ln] = fminf(fmaxf(csum, MINF), 0.f);

    float adj[Ln];
#pragma unroll
    for (int mm = 0; mm < Ln; ++mm) {
        float mx = MINF;
#pragma unroll
        for (int n = 0; n <= Ln; ++n) mx = fmaxf(mx, llg[n] + ndl[n * 11 + mm]);
        float s = 0.f;
#pragma unroll
        for (int n = 0; n <= Ln; ++n) s += expf(llg[n] + ndl[n * 11 + mm] - mx);
        adj[mm] = mx + logf(s);
    }
    float eos_adj[Ln];
    eos_adj[0] = adj[0];
    float accL = log1pf(-expf(eos_adj[0]));
#pragma unroll
    for (int i = 1; i < Ln; ++i) {
        float ai = fminf(fmaxf(adj[i] - accL, MINF), 0.f);
        eos_adj[i] = ai;
        accL += log1pf(-expf(ai));
    }

    // ---- adjusted output: [eos_adj | log_softmax(expected) + log1p(-exp(eos_adj))] ----
    const size_t sec = (size_t)Bn * Ln * Vn;
    float* outAdj = out + sec + (size_t)b * Ln * Vn;
#pragma unroll
    for (int i = 0; i < Ln; ++i) {
        float e = (lane < 31) ? Ex[i][lane] : MINF;
        float mx = wave_max(e);
        float s = wave_sum((lane < 31) ? expf(e - mx) : 0.f);
        float lseR = mx + logf(s);
        float l1me = log1pf(-expf(eos_adj[i]));
        float exm1 = Ex[i][(lane == 0) ? 0 : (lane - 1)];
        float o = (lane == 0) ? eos_adj[i] : (exm1 - lseR + l1me);
        outAdj[i * Vn + lane] = o;
    }

    // ---- noisy: row permutation from mask + EOS tail ----
    const uint8_t* mk = mask + b * Ln;
    int bits[Ln], ndel = 0;
#pragma unroll
    for (int i = 0; i < Ln; ++i) { bits[i] = mk[i] ? 1 : 0; ndel += bits[i]; }
    int rows[Ln];
#pragma unroll
    for (int i = 0; i < Ln; ++i) rows[i] = i;
#pragma unroll
    for (int step = 1; step < Ln; ++step) {
        const int idx = Ln - 1 - step;
        if (bits[idx]) {
            int t = rows[idx];
#pragma unroll
            for (int k = idx; k < Ln - 1; ++k) rows[k] = rows[k + 1];
            rows[Ln - 1] = t;
        }
    }
    float* outNoisy = out + (size_t)b * Ln * Vn;
#pragma unroll
    for (int i = 0; i < Ln; ++i) {
        float v;
        if (i >= Ln - ndel) v = (lane == 0) ? 1.f : 0.f;
        else v = ms[rows[i] * Vn + lane];
        outNoisy[i * Vn + lane] = v;
    }

    // ---- passthrough copies ----
    float* outMsg = out + 2 * sec + (size_t)b * Ln * Vn;
    float* outLg  = out + 3 * sec + (size_t)b * Ln * Vn;
#pragma unroll
    for (int i = 0; i < Ln; ++i) {
        outMsg[i * Vn + lane] = ms[i * Vn + lane];
        outLg[i * Vn + lane] = vals[i];
    }
}

extern "C" void kernel_launch(void* const* d_in, const int* in_sizes, int n_in,
                              void* d_out, int out_size, void* d_ws, size_t ws_size,
                              hipStream_t stream) {
    (void)in_sizes; (void)n_in; (void)out_size; (void)ws_size;
    const float* messages = (const float*)d_in[0];
    const float* logits = (const float*)d_in[1];
    const uint8_t* mask = (const uint8_t*)d_in[2];  // jax bool_ = 1 byte
    float* out = (float*)d_out;
    float* ws = (float*)d_ws;  // needs 313 floats

    dc_init_consts<<<1, 32, 0, stream>>>(ws);
    dc_main<<<Bn, 32, 0, stream>>>(messages, logits, mask, out, ws);
}